// SingleKeyAttention_56487409877249
// MI455X (gfx1250) — compile-verified
//
#include <hip/hip_runtime.h>
#include <hip/hip_bf16.h>

// ---------- problem constants ----------
#define BB    8
#define DD    256
#define HH    128
#define WW    128
#define NCLS  6
#define KK    8
#define CC    48          // NCLS*KK
#define PP    819
#define HW    16384       // HH*WW

// ---------- output layout (flat concat, return order) ----------
#define OFF_ASSPW  0ULL                         // B*D*H*W  = 33554432
#define OFF_PROTO  33554432ULL                  // 48*B*128 = 49152
#define OFF_WMAX   33603584ULL                  // B*H*W    = 131072
#define OFF_SIM    33734656ULL                  // B*48*H*W = 6291456
#define OFF_WMEAN  40026112ULL                  // B*H*W

// ---------- WMMA f32 16x16x4 ----------
typedef __attribute__((ext_vector_type(2))) float v2f;
typedef __attribute__((ext_vector_type(8))) float v8f;

static __device__ inline v8f wmma_f32_16x16x4(v2f a, v2f b, v8f c) {
  // 8 args: (neg_a, A, neg_b, B, c_mod, C, reuse_a, reuse_b)
  return __builtin_amdgcn_wmma_f32_16x16x4_f32(false, a, false, b, (short)0, c,
                                               false, false);
}

// ==========================================================================
// Kernel 1: q[b,c,o] = Wq[i,o,:]·proto[b,c,:] + bq[i,o]   (c = i*8+j)
//           also emits prototypes_out[c,b,o] = q[b,c,o]
// ==========================================================================
__global__ void qproj_kernel(const float* __restrict__ protos,
                             const float* __restrict__ Wq,
                             const float* __restrict__ bq,
                             float* __restrict__ qws,
                             float* __restrict__ out) {
  int id = blockIdx.x * 256 + threadIdx.x;
  if (id >= BB * CC * 128) return;
  int o = id & 127;
  int c = (id >> 7) % CC;
  int b = id / (CC * 128);
  int i = c >> 3;
  const float* wrow = Wq + ((size_t)(i * 128 + o)) * DD;
  const float* prow = protos + ((size_t)(b * CC + c)) * DD;
  float acc = bq[i * 128 + o];
  for (int d = 0; d < DD; ++d) acc += wrow[d] * prow[d];
  qws[((size_t)(b * CC + c)) * 128 + o] = acc;
  out[OFF_PROTO + ((size_t)c * BB + b) * 128 + o] = acc;
}

// ==========================================================================
// Kernel 2: M[b,c,d] = sum_o q[b,c,o] * Wk[o,d]
// ==========================================================================
__global__ void mproj_kernel(const float* __restrict__ qws,
                             const float* __restrict__ Wk,
                             float* __restrict__ Mws) {
  int id = blockIdx.x * 256 + threadIdx.x;
  if (id >= BB * CC * DD) return;
  int d = id & 255;
  int bc = id >> 8;
  const float* qr = qws + (size_t)bc * 128;
  float acc = 0.f;
  for (int o = 0; o < 128; ++o) acc += qr[o] * Wk[o * DD + d];
  Mws[id] = acc;                       // layout ((b*48+c)*256+d)
}

// ==========================================================================
// Kernel 3: sbias[b,c] = sum_o q[b,c,o] * bk[o]
// ==========================================================================
__global__ void sbias_kernel(const float* __restrict__ qws,
                             const float* __restrict__ bk,
                             float* __restrict__ sb) {
  int id = blockIdx.x * 64 + threadIdx.x;
  if (id >= BB * CC) return;
  const float* qr = qws + (size_t)id * 128;
  float acc = 0.f;
  for (int o = 0; o < 128; ++o) acc += qr[o] * bk[o];
  sb[id] = acc;
}

// ==========================================================================
// Kernel 4a/4b: pixel -> sampled-index map
// ==========================================================================
__global__ void map_init(int* __restrict__ map) {
  int i = blockIdx.x * 256 + threadIdx.x;
  if (i < HW) map[i] = -1;
}
__global__ void map_scatter(const int* __restrict__ xs,
                            const int* __restrict__ ys,
                            int* __restrict__ map) {
  int p = blockIdx.x * 256 + threadIdx.x;
  if (p < PP) map[xs[p] * WW + ys[p]] = p;
}

// ==========================================================================
// Kernel 5: argmin over 48 prototypes, one wave32 per (b,p)
//   d2 = ||pf||^2 - 2 sel·pf  ==  sum_d pf*(pf - 2*sel)
// ==========================================================================
__global__ __launch_bounds__(256)
void argmin_kernel(const float* __restrict__ assp,
                   const float* __restrict__ protos,
                   const int* __restrict__ xs,
                   const int* __restrict__ ys,
                   int* __restrict__ idxOut) {
  int wid = blockIdx.x * 8 + (threadIdx.x >> 5);
  int lane = threadIdx.x & 31;
  if (wid >= BB * PP) return;
  int b = wid / PP, p = wid % PP;
  int pix = xs[p] * WW + ys[p];
  float sel[8];
#pragma unroll
  for (int r = 0; r < 8; ++r) {
    int d = lane + 32 * r;
    sel[r] = assp[((size_t)(b * DD + d)) * HW + pix];
  }
  float best = 1e30f;
  int bestc = 0;
  for (int c = 0; c < CC; ++c) {
    const float* pf = protos + ((size_t)(b * CC + c)) * DD;
    float acc = 0.f;
#pragma unroll
    for (int r = 0; r < 8; ++r) {
      float pv = pf[lane + 32 * r];
      acc += pv * (pv - 2.0f * sel[r]);
    }
#pragma unroll
    for (int off = 16; off > 0; off >>= 1)
      acc += __shfl_xor(acc, off, 32);
    if (acc < best) { best = acc; bestc = c; }   // strict < => first min (argmin)
  }
  if (lane == 0) idxOut[wid] = bestc;
}

// ==========================================================================
// Kernel 6: fused main — one workgroup per (row h, batch b), 8 waves.
//   LDS: feats[256][132] | sim[48][132] | wmax[128]
// ==========================================================================
#define FPITCH 132
#define SMEM_FLOATS (DD * FPITCH + CC * FPITCH + 128)

__global__ __launch_bounds__(256)
void fused_main(const float* __restrict__ assp,
                const float* __restrict__ protos,
                const float* __restrict__ Mws,
                const float* __restrict__ sbias,
                const int* __restrict__ map,
                const int* __restrict__ idxbuf,
                float* __restrict__ out) {
  extern __shared__ float smem[];
  float* featsL = smem;                        // 256 x 132
  float* simL   = smem + DD * FPITCH;          // 48 x 132
  float* wmaxL  = simL + CC * FPITCH;          // 128

  const int h = blockIdx.x;
  const int b = blockIdx.y;
  const int t = threadIdx.x;

  // ---- stage 1: async DMA feats tile -> LDS (16B/lane, ASYNCcnt) ----
  {
    const float* gbase = assp + ((size_t)b * DD) * HW + (size_t)h * WW;
    const int dsub = t >> 5;           // 0..7
    const int c4 = (t & 31) * 4;       // column (floats), 16B aligned
#pragma unroll 4
    for (int k = 0; k < 32; ++k) {
      int d = k * 8 + dsub;
      unsigned ldsA = (unsigned)(uintptr_t)(featsL + d * FPITCH + c4);
      const float* gp = gbase + (size_t)d * HW + c4;
      asm volatile("global_load_async_to_lds_b128 %0, %1, off"
                   :: "v"(ldsA), "v"(gp)
                   : "memory");
    }
  }

  // replacement index for this row's pixels (overlaps the DMA)
  int rc = -1;
  if (t < 128) {
    int p = map[h * WW + t];
    rc = (p >= 0) ? idxbuf[b * PP + p] : -1;
  }

  asm volatile("s_wait_asynccnt 0x0" ::: "memory");
  __syncthreads();

  // fix-up: overwrite sampled pixels' feature column with nearest prototype
  if (rc >= 0) {
    const float* prow = protos + ((size_t)(b * CC + rc)) * DD;
    for (int d = 0; d < DD; ++d) featsL[d * FPITCH + t] = prow[d];
  }
  __syncthreads();

  // ---- stage 2: sim[48,128] = M[b] (48x256) x featsL (256x128) via WMMA ----
  const int lane = t & 31;
  const int wv = t >> 5;                       // 0..7 -> pixel block
  const int pixBase = wv * 16;
  const int mrow = lane & 15;
  const int khalf = (lane >> 4) * 2;           // 0 or 2

  v8f acc[3];
#pragma unroll
  for (int cb = 0; cb < 3; ++cb) { v8f z = {}; acc[cb] = z; }

  const float* Mb = Mws + (size_t)b * CC * DD;
  for (int kk = 0; kk < 64; ++kk) {
    int dB = kk * 4 + khalf;
    v2f bv;
    bv.x = featsL[dB * FPITCH + pixBase + mrow];
    bv.y = featsL[(dB + 1) * FPITCH + pixBase + mrow];
#pragma unroll
    for (int cb = 0; cb < 3; ++cb) {
      const float* ar = Mb + (size_t)(cb * 16 + mrow) * DD + dB;
      v2f av;
      av.x = ar[0];
      av.y = ar[1];
      acc[cb] = wmma_f32_16x16x4(av, bv, acc[cb]);
    }
  }

  // add sbias, deposit sim tile in LDS
  {
    int col = pixBase + mrow;
    int rofs = (lane >= 16) ? 8 : 0;
#pragma unroll
    for (int cb = 0; cb < 3; ++cb) {
#pragma unroll
      for (int r = 0; r < 8; ++r) {
        int crow = cb * 16 + r + rofs;
        simL[crow * FPITCH + col] = acc[cb][r] + sbias[b * CC + crow];
      }
    }
  }
  __syncthreads();

  // ---- async DMA simL -> out.sim (overlaps softmax + weighted writes);
  //      S_ENDPGM's implicit wait-idle covers completion ----
  {
    float* sbase = out + OFF_SIM + ((size_t)b * CC) * HW + (size_t)h * WW;
#pragma unroll
    for (int k = 0; k < 6; ++k) {
      int idx = k * 256 + t;           // 0..1535 float4 chunks
      int row = idx >> 5;              // 0..47
      int c4 = (idx & 31) * 4;
      unsigned ldsA = (unsigned)(uintptr_t)(simL + row * FPITCH + c4);
      float* gp = sbase + (size_t)row * HW + c4;
      asm volatile("global_store_async_from_lds_b128 %0, %1, off"
                   :: "v"(gp), "v"(ldsA)
                   : "memory");
    }
  }

  // ---- stage 3: per-pixel softmax stats over 48 classes ----
  // s_c = sim_c/6 ; w_max = exp(0)/Z = 1/Z ; mean of softmax = 1/48 exactly
  if (t < 128) {
    float m = -1e30f;
    for (int c = 0; c < CC; ++c) m = fmaxf(m, simL[c * FPITCH + t]);
    float Z = 0.f;
    for (int c = 0; c < CC; ++c)
      Z += __expf((simL[c * FPITCH + t] - m) * (1.0f / 6.0f));
    float wm = 1.0f / Z;
    wmaxL[t] = wm;
    size_t o = (size_t)b * HW + (size_t)h * WW + t;
    out[OFF_WMAX + o] = wm;
    out[OFF_WMEAN + o] = 1.0f / 48.0f;
  }
  __syncthreads();

  // ---- stage 4: assp_weighted = feats * w_max (float4, b128 path) ----
  {
    const int dsub = t >> 5;
    const int c4 = (t & 31) * 4;
    float4 wv4 = *(const float4*)(wmaxL + c4);
    float* obase = out + OFF_ASSPW + ((size_t)b * DD) * HW + (size_t)h * WW;
#pragma unroll 4
    for (int k = 0; k < 32; ++k) {
      int d = k * 8 + dsub;
      float4 f4 = *(const float4*)(featsL + d * FPITCH + c4);
      float4 o4;
      o4.x = f4.x * wv4.x;
      o4.y = f4.y * wv4.y;
      o4.z = f4.z * wv4.z;
      o4.w = f4.w * wv4.w;
      *(float4*)(obase + (size_t)d * HW + c4) = o4;
    }
  }
}

// ==========================================================================
// Launch
// ==========================================================================
extern "C" void kernel_launch(void* const* d_in, const int* in_sizes, int n_in,
                              void* d_out, int out_size, void* d_ws, size_t ws_size,
                              hipStream_t stream) {
  const float* assp   = (const float*)d_in[0];
  const float* protos = (const float*)d_in[1];
  const float* Wk     = (const float*)d_in[2];
  const float* bk     = (const float*)d_in[3];
  const float* Wq     = (const float*)d_in[4];
  const float* bq     = (const float*)d_in[5];
  const int*   xs     = (const int*)d_in[6];
  const int*   ys     = (const int*)d_in[7];
  float* out = (float*)d_out;

  // workspace partition
  char* ws = (char*)d_ws;
  float* qws   = (float*)ws;                               // 8*48*128
  float* Mws   = (float*)(ws + 196608);                    // 8*48*256
  float* sbias = (float*)(ws + 196608 + 393216);           // 8*48
  int*   idxb  = (int*)  (ws + 196608 + 393216 + 2048);    // 8*819
  int*   map   = (int*)  (ws + 196608 + 393216 + 2048 + 32768); // 16384

  const int smemBytes = SMEM_FLOATS * sizeof(float);       // ~157 KB
  (void)hipFuncSetAttribute((const void*)fused_main,
                            hipFuncAttributeMaxDynamicSharedMemorySize,
                            smemBytes);

  qproj_kernel<<<(BB * CC * 128 + 255) / 256, 256, 0, stream>>>(protos, Wq, bq, qws, out);
  mproj_kernel<<<(BB * CC * DD + 255) / 256, 256, 0, stream>>>(qws, Wk, Mws);
  sbias_kernel<<<(BB * CC + 63) / 64, 64, 0, stream>>>(qws, bk, sbias);
  map_init<<<HW / 256, 256, 0, stream>>>(map);
  map_scatter<<<(PP + 255) / 256, 256, 0, stream>>>(xs, ys, map);
  argmin_kernel<<<(BB * PP + 7) / 8, 256, 0, stream>>>(assp, protos, xs, ys, idxb);

  dim3 grid(HH, BB);
  fused_main<<<grid, 256, smemBytes, stream>>>(assp, protos, Mws, sbias, map, idxb, out);
}